// AttnPool_73100343378373
// MI455X (gfx1250) — compile-verified
//
#include <hip/hip_runtime.h>
#include <hip/hip_bf16.h>

typedef __attribute__((ext_vector_type(2))) float v2f;
typedef __attribute__((ext_vector_type(8))) float v8f;

#define B_ 16
#define L_ 4096
#define D_ 1024
#define H_ 8

// ---------------------------------------------------------------------------
// Kernel 1: qwH[h*D + f] = sum_g q[h,g] * w[g,f]  for h<8; rows 8..15 zeroed.
// Head-major layout so the WMMA B fragments in the scores kernel are
// contiguous b64 loads. 64 KB total -> L2 resident for the whole scores pass.
// ---------------------------------------------------------------------------
__global__ void qw_kernel(const float* __restrict__ q, const float* __restrict__ w,
                          float* __restrict__ qwH) {
  int f = blockIdx.x * blockDim.x + threadIdx.x;  // 0..1023
  float acc[H_];
#pragma unroll
  for (int h = 0; h < H_; ++h) acc[h] = 0.f;
  for (int g = 0; g < D_; ++g) {
    float wv = w[g * D_ + f];              // coalesced across threads
#pragma unroll
    for (int h = 0; h < H_; ++h) acc[h] += q[h * D_ + g] * wv;  // scalar-broadcast q
  }
#pragma unroll
  for (int h = 0; h < H_; ++h) qwH[h * D_ + f] = acc[h];
#pragma unroll
  for (int h = H_; h < 16; ++h) qwH[h * D_ + f] = 0.f;          // pad heads
}

// ---------------------------------------------------------------------------
// Kernel 2: scores via V_WMMA_F32_16X16X4_F32.
// One wave per 16 rows of X (rows never straddle a batch: 4096 % 16 == 0).
// D = A(16 rows x 4 f) * B(4 f x 16 heads) accumulated over f = 0..1023.
// A lane j: row j&15, K-pair 2*(j>>4)     -> contiguous b64 from x
// B lane j: head j&15, same K-pair        -> contiguous b64 from qwH
// After the K loop, apply the key-padding mask and scatter into the
// (b, head, l) padded attention buffer.
// ---------------------------------------------------------------------------
__global__ void scores_kernel(const float* __restrict__ x,
                              const unsigned char* __restrict__ kpm,
                              const float* __restrict__ qwH,
                              float* __restrict__ attnP) {
  int wave = (blockIdx.x * blockDim.x + threadIdx.x) >> 5;
  int lane = threadIdx.x & 31;
  int r0   = wave * 16;                 // first global row (b*L + l)
  int m    = lane & 15;                 // A row / B head column
  int koff = (lane >> 4) << 1;          // K sub-offset {0,2}

  const float* arow = x   + (size_t)(r0 + m) * D_ + koff;
  const float* brow = qwH + (size_t)m        * D_ + koff;

  v8f c = {};
#pragma unroll 4
  for (int f0 = 0; f0 < D_; f0 += 4) {
    v2f a = *(const v2f*)(arow + f0);
    v2f b = *(const v2f*)(brow + f0);
    c = __builtin_amdgcn_wmma_f32_16x16x4_f32(false, a, false, b,
                                              (short)0, c, false, false);
  }

  int head = m;                         // C col = lane&15
  if (head < H_) {
    int rbase = r0 + ((lane >> 4) << 3);  // C row = v + 8*(lane>>4)
#pragma unroll
    for (int v = 0; v < 8; ++v) {
      int r = rbase + v;
      int b = r >> 12;                  // r / 4096
      int l = r & (L_ - 1);
      float val = kpm[r] ? -__builtin_inff() : c[v];
      attnP[((size_t)(b * 16 + head)) * L_ + l] = val;
    }
  }
}

// ---------------------------------------------------------------------------
// Kernel 3: softmax over L per (b, head) row, in place in attnP.
// One 256-thread block per padded row; pad rows (h >= 8) are zero-filled so
// the pooling pass reads clean data. Values held in registers -> one global
// read + one global write per element.
// ---------------------------------------------------------------------------
__global__ void softmax_kernel(float* __restrict__ attnP) {
  int row = blockIdx.x;                 // b*16 + h
  int h   = row & 15;
  int tid = threadIdx.x;
  float* p = attnP + (size_t)row * L_;

  if (h >= H_) {                        // uniform per block
    for (int i = tid; i < L_; i += 256) p[i] = 0.f;
    return;
  }

  __shared__ float red[256];
  float v[L_ / 256];
  float m = -__builtin_inff();
#pragma unroll
  for (int i = 0; i < L_ / 256; ++i) { v[i] = p[tid + i * 256]; m = fmaxf(m, v[i]); }
  red[tid] = m; __syncthreads();
  for (int s = 128; s > 0; s >>= 1) {
    if (tid < s) red[tid] = fmaxf(red[tid], red[tid + s]);
    __syncthreads();
  }
  float rowmax = red[0]; __syncthreads();

  float sum = 0.f;
#pragma unroll
  for (int i = 0; i < L_ / 256; ++i) { v[i] = expf(v[i] - rowmax); sum += v[i]; }
  red[tid] = sum; __syncthreads();
  for (int s = 128; s > 0; s >>= 1) {
    if (tid < s) red[tid] += red[tid + s];
    __syncthreads();
  }
  float inv = 1.f / red[0];
#pragma unroll
  for (int i = 0; i < L_ / 256; ++i) p[tid + i * 256] = v[i] * inv;
}

// ---------------------------------------------------------------------------
// Kernel 4: pooled_b = attn_b(16pad x 4096) @ x_b(4096 x 1024), WMMA f32.
// One wave per (b, 16-column tile): 16 b * 64 tiles = 1024 waves,
// K loop over L in steps of 4 -> 1024 WMMAs per wave.
// A lane j: head j&15, K-pair 2*(j>>4)  -> contiguous b64 from attnP
// B lane j: col n0+(j&15), rows l0+koff+{0,1} -> two b32 loads (64B/half-wave)
// C rows = heads; lanes 0..15 hold heads 0..7 -> write those to d_out.
// ---------------------------------------------------------------------------
__global__ void pool_kernel(const float* __restrict__ x,
                            const float* __restrict__ attnP,
                            float* __restrict__ out) {
  int wave = (blockIdx.x * blockDim.x + threadIdx.x) >> 5;
  int lane = threadIdx.x & 31;
  int b    = wave >> 6;
  int n0   = (wave & 63) << 4;
  int m    = lane & 15;
  int koff = (lane >> 4) << 1;

  const float* arow = attnP + (size_t)(b * 16 + m) * L_ + koff;
  const float* bcol = x + (size_t)b * L_ * D_ + (size_t)koff * D_ + n0 + m;

  v8f c = {};
#pragma unroll 4
  for (int l0 = 0; l0 < L_; l0 += 4) {
    v2f a = *(const v2f*)(arow + l0);
    v2f bb;
    bb.x = bcol[(size_t)l0 * D_];
    bb.y = bcol[(size_t)l0 * D_ + D_];
    c = __builtin_amdgcn_wmma_f32_16x16x4_f32(false, a, false, bb,
                                              (short)0, c, false, false);
  }

  if (lane < 16) {                      // heads 0..7 live in lanes 0..15
#pragma unroll
    for (int v = 0; v < 8; ++v) {
      out[(size_t)b * (H_ * D_) + (size_t)v * D_ + n0 + m] = c[v];
    }
  }
}

// ---------------------------------------------------------------------------
extern "C" void kernel_launch(void* const* d_in, const int* in_sizes, int n_in,
                              void* d_out, int out_size, void* d_ws, size_t ws_size,
                              hipStream_t stream) {
  const float*         x   = (const float*)d_in[0];
  const unsigned char* kpm = (const unsigned char*)d_in[1];  // bool -> 1 byte
  const float*         q   = (const float*)d_in[2];
  const float*         w   = (const float*)d_in[3];
  float* out = (float*)d_out;

  // Workspace: qwH (16*1024 f32 = 64 KB) | attnP (16*16*4096 f32 = 4 MB)
  float* qwH   = (float*)d_ws;
  float* attnP = qwH + 16 * 1024;

  qw_kernel     <<<D_ / 256,            256, 0, stream>>>(q, w, qwH);
  scores_kernel <<<(B_ * L_ / 16) / 4,  128, 0, stream>>>(x, kpm, qwH, attnP);
  softmax_kernel<<<B_ * 16,             256, 0, stream>>>(attnP);
  pool_kernel   <<<(B_ * 64) / 4,       128, 0, stream>>>(x, attnP, out);
}